// SVD_Trunc_8572754723275
// MI455X (gfx1250) — compile-verified
//
#include <hip/hip_runtime.h>
#include <hip/hip_bf16.h>
#include <math.h>

// ---------------------------------------------------------------------------
// Pipeline: 1x1-conv GEMM -> BN+ReLU -> row-center -> batched covariance ->
// Newton-Schulz matrix sqrt (replaces fp64 eigh) -> upper-triangle extract.
// All GEMMs use CDNA5 V_WMMA_F32_16X16X4_F32 (wave32, fp32 matrix path),
// with 2x2 (NS/cov) or 2x1 (conv) register blocking per wave so each WMMA
// costs only ~8 B/lane of operand traffic.
// ---------------------------------------------------------------------------

typedef __attribute__((ext_vector_type(2))) float v2f;
typedef __attribute__((ext_vector_type(8))) float v8f;

#define BATCH 64
#define CIN   2048
#define DCH   256     // output channels / matrix dim
#define HW    196     // 14*14
#define MP    208     // HW padded to multiple of 16 (zeros -> cov unchanged)
#define NTRI  32896   // 256*257/2

// WMMA fp32 16x16x4.  Lane l: m/n = l&15, half = l>>4.
//   A frag: a.x = A[m][k+2*half], a.y = A[m][k+2*half+1]
//   B frag: b.x = B[k+2*half][n], b.y = B[k+2*half+1][n]
//   D frag: acc[r] = D[r + 8*half][n]
__device__ __forceinline__ v8f wmma_f32(v2f a, v2f b, v8f c) {
  return __builtin_amdgcn_wmma_f32_16x16x4_f32(
      false, a, false, b, (short)0, c, false, false);
}

// ---- K1: y[b][o][hw] = sum_c W[o][c] * x[b][c][hw] ------------------------
// Wave computes a 32x16 block (2 o-tiles x 1 n-tile): the strided x fragment
// is loaded once and reused by both WMMAs.  Branch-free inner loop: columns
// >= 196 use a clamped address and a 0.0 multiplicative mask.
__global__ void conv_gemm_k(const float* __restrict__ W,
                            const float* __restrict__ x,
                            float* __restrict__ y) {
  const int b    = blockIdx.x;
  const int wave = threadIdx.x >> 5;
  const int id   = blockIdx.y * 4 + wave;       // 0..103
  const int oi   = id / 13;                     // 8 o-tile *pairs*
  const int nj   = id % 13;                     // 13 n-tiles
  const int l    = threadIdx.x & 31;
  const int m    = l & 15;
  const int half = l >> 4;
  const int col  = nj * 16 + m;
  const bool valid = (col < HW);
  const int  cols  = valid ? col : 0;           // safe address
  const float bmask = valid ? 1.0f : 0.0f;      // zero the pad lanes

  const float* wrow0 = W + (size_t)(oi * 32 + m) * CIN;
  const float* wrow1 = wrow0 + (size_t)16 * CIN;
  const float* xb    = x + (size_t)b * CIN * HW + cols;

  v8f acc0 = {}, acc1 = {};
#pragma unroll 2
  for (int k = 0; k < CIN; k += 4) {
    const int kk = k + 2 * half;
    v2f a0 = *(const v2f*)(wrow0 + kk);
    v2f a1 = *(const v2f*)(wrow1 + kk);
    v2f bv;
    bv.x = bmask * xb[(size_t)kk * HW];
    bv.y = bmask * xb[(size_t)(kk + 1) * HW];
    acc0 = wmma_f32(a0, bv, acc0);
    acc1 = wmma_f32(a1, bv, acc1);
  }

  if (valid) {
    float* yb = y + ((size_t)b * DCH + (size_t)oi * 32) * HW + col;
#pragma unroll
    for (int r = 0; r < 8; ++r) {
      yb[(size_t)(r + 8 * half) * HW]        = acc0[r];
      yb[(size_t)(r + 8 * half + 16) * HW]   = acc1[r];
    }
  }
}

// ---- K2: per-channel BN statistics over (B,H,W) ---------------------------
__global__ void bn_stats_k(const float* __restrict__ y,
                           const float* __restrict__ gamma,
                           const float* __restrict__ beta,
                           float* __restrict__ scale,
                           float* __restrict__ shift) {
  const int o = blockIdx.x, tid = threadIdx.x;
  __shared__ float ssum[256], ssq[256];
  float s = 0.0f, q = 0.0f;
  for (int idx = tid; idx < BATCH * HW; idx += 256) {
    const int b = idx / HW, hw = idx % HW;
    const float v = y[((size_t)b * DCH + o) * HW + hw];
    s += v; q += v * v;
  }
  ssum[tid] = s; ssq[tid] = q;
  __syncthreads();
  for (int off = 128; off > 0; off >>= 1) {
    if (tid < off) { ssum[tid] += ssum[tid + off]; ssq[tid] += ssq[tid + off]; }
    __syncthreads();
  }
  if (tid == 0) {
    const float n    = (float)(BATCH * HW);
    const float mean = ssum[0] / n;
    const float var  = ssq[0] / n - mean * mean;
    const float sc   = gamma[o] * rsqrtf(var + 1e-5f);
    scale[o] = sc;
    shift[o] = beta[o] - mean * sc;
  }
}

// ---- K3: BN+ReLU, subtract per-(b,o) row mean, write zero-padded 208 cols -
__global__ void center_rows_k(const float* __restrict__ y,
                              const float* __restrict__ scale,
                              const float* __restrict__ shift,
                              float* __restrict__ zc) {
  const int o = blockIdx.x, b = blockIdx.y, tid = threadIdx.x;
  __shared__ float ssum[256];
  float v = 0.0f;
  if (tid < HW)
    v = fmaxf(scale[o] * y[((size_t)b * DCH + o) * HW + tid] + shift[o], 0.0f);
  ssum[tid] = v;
  __syncthreads();
  for (int off = 128; off > 0; off >>= 1) {
    if (tid < off) ssum[tid] += ssum[tid + off];
    __syncthreads();
  }
  const float mean = ssum[0] * (1.0f / (float)HW);
  if (tid < MP)
    zc[((size_t)b * DCH + o) * MP + tid] = (tid < HW) ? (v - mean) : 0.0f;
}

// ---- K4: cov_b = zc_b @ zc_b^T / HW  ------------------------------------
// Wave computes a 32x32 block (2x2 tiles); B operand = rows of zc so all
// fragment loads are contiguous 8-byte v2f.
__global__ void cov_gemm_k(const float* __restrict__ zc,
                           float* __restrict__ cov) {
  const int b    = blockIdx.x;
  const int wave = threadIdx.x >> 5;
  const int id   = blockIdx.y * 4 + wave;       // 0..63
  const int bi   = id >> 3, bj = id & 7;        // 8x8 blocks of 32x32
  const int l    = threadIdx.x & 31;
  const int m    = l & 15;
  const int half = l >> 4;

  const float* zb = zc + (size_t)b * DCH * MP;
  const float* a0 = zb + (size_t)(bi * 32 + m) * MP;
  const float* a1 = a0 + (size_t)16 * MP;
  const float* b0 = zb + (size_t)(bj * 32 + m) * MP;   // B[k][n] = zc[n][k]
  const float* b1 = b0 + (size_t)16 * MP;

  v8f acc00 = {}, acc01 = {}, acc10 = {}, acc11 = {};
#pragma unroll 2
  for (int k = 0; k < MP; k += 4) {
    const int kk = k + 2 * half;
    v2f fa0 = *(const v2f*)(a0 + kk);
    v2f fa1 = *(const v2f*)(a1 + kk);
    v2f fb0 = *(const v2f*)(b0 + kk);
    v2f fb1 = *(const v2f*)(b1 + kk);
    acc00 = wmma_f32(fa0, fb0, acc00);
    acc01 = wmma_f32(fa0, fb1, acc01);
    acc10 = wmma_f32(fa1, fb0, acc10);
    acc11 = wmma_f32(fa1, fb1, acc11);
  }

  float* cb = cov + (size_t)b * DCH * DCH + (size_t)(bi * 32) * DCH + bj * 32 + m;
  const float inv = 1.0f / (float)HW;
#pragma unroll
  for (int r = 0; r < 8; ++r) {
    const size_t row = (size_t)(r + 8 * half) * DCH;
    cb[row]                       = acc00[r] * inv;
    cb[row + 16]                  = acc01[r] * inv;
    cb[row + (size_t)16 * DCH]    = acc10[r] * inv;
    cb[row + (size_t)16 * DCH+16] = acc11[r] * inv;
  }
}

// ---- K5: trace, Y0 = cov/tr, Z0 = I ---------------------------------------
__global__ void trace_init_k(const float* __restrict__ cov,
                             float* __restrict__ Y, float* __restrict__ Z,
                             float* __restrict__ s) {
  const int b = blockIdx.x, tid = threadIdx.x;
  __shared__ float sh[256];
  const float* cb = cov + (size_t)b * DCH * DCH;
  sh[tid] = cb[(size_t)tid * DCH + tid];
  __syncthreads();
  for (int off = 128; off > 0; off >>= 1) {
    if (tid < off) sh[tid] += sh[tid + off];
    __syncthreads();
  }
  const float tr = fmaxf(sh[0], 1e-30f);
  if (tid == 0) s[b] = tr;
  const float inv = 1.0f / tr;
  float* Yb = Y + (size_t)b * DCH * DCH;
  float* Zb = Z + (size_t)b * DCH * DCH;
  for (int e = tid; e < DCH * DCH; e += 256) {
    Yb[e] = cb[e] * inv;
    Zb[e] = ((e >> 8) == (e & 255)) ? 1.0f : 0.0f;
  }
}

// ---- K6: batched C = alpha*(A@B) + beta*I, all operands symmetric ---------
// B symmetric => B[k][n] = B[n][k]: contiguous v2f fragment loads; 2x2
// register blocking => 4 WMMAs per 4 loads.
__global__ void ns_gemm_k(const float* __restrict__ A,
                          const float* __restrict__ B,
                          float* __restrict__ C, float alpha, float beta) {
  const int b    = blockIdx.x;
  const int wave = threadIdx.x >> 5;
  const int id   = blockIdx.y * 4 + wave;       // 0..63
  const int bi   = id >> 3, bj = id & 7;
  const int l    = threadIdx.x & 31;
  const int m    = l & 15;
  const int half = l >> 4;

  const float* Ab = A + (size_t)b * DCH * DCH;
  const float* Bb = B + (size_t)b * DCH * DCH;
  const float* a0 = Ab + (size_t)(bi * 32 + m) * DCH;
  const float* a1 = a0 + (size_t)16 * DCH;
  const float* b0 = Bb + (size_t)(bj * 32 + m) * DCH;
  const float* b1 = b0 + (size_t)16 * DCH;

  v8f acc00 = {}, acc01 = {}, acc10 = {}, acc11 = {};
#pragma unroll 2
  for (int k = 0; k < DCH; k += 4) {
    const int kk = k + 2 * half;
    v2f fa0 = *(const v2f*)(a0 + kk);
    v2f fa1 = *(const v2f*)(a1 + kk);
    v2f fb0 = *(const v2f*)(b0 + kk);
    v2f fb1 = *(const v2f*)(b1 + kk);
    acc00 = wmma_f32(fa0, fb0, acc00);
    acc01 = wmma_f32(fa0, fb1, acc01);
    acc10 = wmma_f32(fa1, fb0, acc10);
    acc11 = wmma_f32(fa1, fb1, acc11);
  }

  float* cb = C + (size_t)b * DCH * DCH + (size_t)(bi * 32) * DCH + bj * 32 + m;
#pragma unroll
  for (int r = 0; r < 8; ++r) {
    const int gi0 = bi * 32 + r + 8 * half;
    const int gi1 = gi0 + 16;
    const int gj0 = bj * 32 + m;
    const int gj1 = gj0 + 16;
    const size_t row = (size_t)(r + 8 * half) * DCH;
    cb[row]                        = alpha * acc00[r] + ((gi0 == gj0) ? beta : 0.0f);
    cb[row + 16]                   = alpha * acc01[r] + ((gi0 == gj1) ? beta : 0.0f);
    cb[row + (size_t)16 * DCH]     = alpha * acc10[r] + ((gi1 == gj0) ? beta : 0.0f);
    cb[row + (size_t)16 * DCH+16]  = alpha * acc11[r] + ((gi1 == gj1) ? beta : 0.0f);
  }
}

// ---- K7: Q = sqrt(trace)*Y, extract upper triangle (triu_indices order) ---
__global__ void finalize_k(const float* __restrict__ Y,
                           const float* __restrict__ s,
                           float* __restrict__ out) {
  const int i = blockIdx.x, b = blockIdx.y, j = threadIdx.x;
  if (j < i) return;
  const size_t rowoff = (size_t)i * DCH - (size_t)i * (i - 1) / 2;
  out[(size_t)b * NTRI + rowoff + (size_t)(j - i)] =
      sqrtf(s[b]) * Y[((size_t)b * DCH + i) * DCH + j];
}

extern "C" void kernel_launch(void* const* d_in, const int* in_sizes, int n_in,
                              void* d_out, int out_size, void* d_ws, size_t ws_size,
                              hipStream_t stream) {
  const float* x     = (const float*)d_in[0];   // (64,2048,14,14)
  const float* W     = (const float*)d_in[1];   // (256,2048)
  const float* gamma = (const float*)d_in[2];   // (256,)
  const float* beta  = (const float*)d_in[3];   // (256,)
  float* out = (float*)d_out;

  // Workspace carve-up (floats), ~111 MB total.
  float* ws = (float*)d_ws;
  size_t off = 0;
  float* y     = ws + off; off += (size_t)BATCH * DCH * HW;
  float* scale = ws + off; off += DCH;
  float* shift = ws + off; off += DCH;
  float* zc    = ws + off; off += (size_t)BATCH * DCH * MP;
  float* cov   = ws + off; off += (size_t)BATCH * DCH * DCH;  // reused as T
  float* Yb    = ws + off; off += (size_t)BATCH * DCH * DCH;
  float* Zb    = ws + off; off += (size_t)BATCH * DCH * DCH;
  float* Y2    = ws + off; off += (size_t)BATCH * DCH * DCH;
  float* Z2    = ws + off; off += (size_t)BATCH * DCH * DCH;
  float* sArr  = ws + off; off += BATCH;
  (void)ws_size; (void)in_sizes; (void)n_in; (void)out_size;

  // 1) conv GEMM: 8 o-pairs x 13 n-tiles = 104 waves/batch, 4 waves/block
  conv_gemm_k<<<dim3(BATCH, 26), 128, 0, stream>>>(W, x, y);
  // 2) BN statistics per channel
  bn_stats_k<<<DCH, 256, 0, stream>>>(y, gamma, beta, scale, shift);
  // 3) BN+ReLU + row centering, zero-padded to 208 cols
  center_rows_k<<<dim3(DCH, BATCH), 256, 0, stream>>>(y, scale, shift, zc);
  // 4) batched covariance: 64 waves/batch (8x8 blocks of 32x32)
  cov_gemm_k<<<dim3(BATCH, 16), 128, 0, stream>>>(zc, cov);
  // 5) trace normalize + init
  trace_init_k<<<BATCH, 256, 0, stream>>>(cov, Yb, Zb, sArr);

  // 6) Newton-Schulz coupled iteration:  T = 1.5I - 0.5*Z*Y;  Y<-Y*T;  Z<-T*Z
  float* T = cov;
  float* Ycur = Yb; float* Zcur = Zb; float* Ynxt = Y2; float* Znxt = Z2;
  for (int it = 0; it < 16; ++it) {
    ns_gemm_k<<<dim3(BATCH, 16), 128, 0, stream>>>(Zcur, Ycur, T, -0.5f, 1.5f);
    ns_gemm_k<<<dim3(BATCH, 16), 128, 0, stream>>>(Ycur, T, Ynxt, 1.0f, 0.0f);
    ns_gemm_k<<<dim3(BATCH, 16), 128, 0, stream>>>(T, Zcur, Znxt, 1.0f, 0.0f);
    float* t0 = Ycur; Ycur = Ynxt; Ynxt = t0;
    float* t1 = Zcur; Zcur = Znxt; Znxt = t1;
  }

  // 7) sqrt(cov) = sqrt(trace) * Y; upper-triangle extract
  finalize_k<<<dim3(DCH, BATCH), 256, 0, stream>>>(Ycur, sArr, out);
}